// A_MMMSE_GPU_3161095930400
// MI455X (gfx1250) — compile-verified
//
#include <hip/hip_runtime.h>
#include <hip/hip_bf16.h>
#include <math.h>
#include <stdint.h>

// ---------------------------------------------------------------------------
// WMMSE step for I=65536 users, R=8, T=16, D=4 (complex64 as interleaved f32)
// ---------------------------------------------------------------------------
#define I_USERS 65536
#define R_ANT   8
#define T_ANT   16
#define D_STR   4
#define SIGMA_C 1.0f
#define P_TOT_C 262144.0f
#define ALPHA_C 1.0f
#define LR_C    0.01f

typedef float v2f __attribute__((ext_vector_type(2)));
typedef float v8f __attribute__((ext_vector_type(8)));

// --- CDNA5 async global->LDS path (guarded: falls back to sync loads) -------
#if defined(__has_builtin)
#  if __has_builtin(__builtin_amdgcn_global_load_async_to_lds_b128)
#    define HAVE_ASYNC_LDS 1
#  endif
#endif
#ifndef HAVE_ASYNC_LDS
#  define HAVE_ASYNC_LDS 0
#endif

#if defined(__has_builtin)
#  if __has_builtin(__builtin_amdgcn_s_wait_asynccnt)
#    define WAIT_ASYNC(n) __builtin_amdgcn_s_wait_asynccnt(n)
#  endif
#endif
#ifndef WAIT_ASYNC
#  define WAIT_ASYNC(n) asm volatile("s_wait_asynccnt %0" ::"i"(n) : "memory")
#endif

#if HAVE_ASYNC_LDS
// Builtin signature (from clang diagnostic): param0 = v4i addrspace(1)*,
// param1 = v4i addrspace(3)*, then imm offset, imm cpol.
typedef int v4i_t __attribute__((vector_size(16)));
typedef __attribute__((address_space(1))) v4i_t* gv4_t;   // global
typedef __attribute__((address_space(3))) v4i_t* lv4_t;   // LDS
// Copy one 16-byte chunk per active lane, global -> LDS, async (ASYNCcnt).
__device__ __forceinline__ void async_cp16(void* lds, const void* g) {
  __builtin_amdgcn_global_load_async_to_lds_b128(
      (gv4_t)(uintptr_t)g, (lv4_t)(uintptr_t)lds, 0, 0);
}
#endif

// Workspace layout in floats:
//   [0..255]     VVh_re (16x16 row-major)
//   [256..511]   VVh_im
//   [512..767]   J_re
//   [768..1023]  J_im
//   [1024]       P_sum
//   [2048.. ]    UW per user: I*R*D*2 floats (8x4 complex each)
#define WS_VVH_RE 0
#define WS_VVH_IM 256
#define WS_J_RE   512
#define WS_J_IM   768
#define WS_P      1024
#define WS_UW     2048

// ------------------------- complex helpers (float2) -------------------------
__device__ __forceinline__ float2 cmul(float2 a, float2 b) {
  return make_float2(a.x * b.x - a.y * b.y, a.x * b.y + a.y * b.x);
}
__device__ __forceinline__ float2 csub(float2 a, float2 b) {
  return make_float2(a.x - b.x, a.y - b.y);
}
// c += a*b
__device__ __forceinline__ float2 cfma_(float2 a, float2 b, float2 c) {
  return make_float2(fmaf(a.x, b.x, fmaf(-a.y, b.y, c.x)),
                     fmaf(a.x, b.y, fmaf( a.y, b.x, c.y)));
}
// c += a*conj(b)
__device__ __forceinline__ float2 cfmac(float2 a, float2 b, float2 c) {
  return make_float2(fmaf(a.x, b.x, fmaf( a.y, b.y, c.x)),
                     fmaf(a.y, b.x, fmaf(-a.x, b.y, c.y)));
}
// c += conj(a)*b
__device__ __forceinline__ float2 cfmca(float2 a, float2 b, float2 c) {
  return make_float2(fmaf(a.x, b.x, fmaf( a.y, b.y, c.x)),
                     fmaf(a.x, b.y, fmaf(-a.y, b.x, c.y)));
}
__device__ __forceinline__ float2 crecip(float2 a) {
  float inv = 1.0f / fmaf(a.x, a.x, a.y * a.y);
  return make_float2(a.x * inv, -a.y * inv);
}

// CDNA5 f32 WMMA: D(16x16) = A(16x4) * B(4x16) + C
__device__ __forceinline__ v8f wmma4(v2f a, v2f b, v8f c) {
  return __builtin_amdgcn_wmma_f32_16x16x4_f32(false, a, false, b,
                                               (short)0, c, false, false);
}

// ------------------------------- kernels -------------------------------------
__global__ void k_init(float* __restrict__ ws) {
  int i = blockIdx.x * blockDim.x + threadIdx.x;
  if (i < 2048) ws[i] = 0.0f;
}

// VVh = sum_i V_i V_i^H  via WMMA rank-4 outer products (complex -> 4 real WMMAs)
__global__ void k_vvh(const float* __restrict__ Vg, float* __restrict__ ws) {
  const int lane = threadIdx.x;            // wave32, one wave per block
  const int m  = lane & 15;                // row (A) / col (B)
  const int kk = lane >> 4;                // half-wave selector
  v8f cre = {0.f,0.f,0.f,0.f,0.f,0.f,0.f,0.f};
  v8f cim = {0.f,0.f,0.f,0.f,0.f,0.f,0.f,0.f};
  for (int u = blockIdx.x; u < I_USERS; u += gridDim.x) {
    const float* vb = Vg + (size_t)u * (T_ANT * D_STR * 2);
    const int un = u + (int)gridDim.x;
    if (un < I_USERS)   // locality 3 -> WGP scope: pull into all cache levels
      __builtin_prefetch(Vg + (size_t)un * (T_ANT * D_STR * 2), 0, 3);
    // A = V (16x4): lane half 0 -> K=0,1 ; half 1 -> K=2,3
    const int k0 = kk * 2;
    float4 av = *(const float4*)(vb + (m * D_STR + k0) * 2);
    v2f Ar = { av.x, av.z };
    v2f Ai = { av.y, av.w };
    // B = V^H (4x16): VGPR0 row kk, VGPR1 row kk+2, column n = m
    float2 b0 = *(const float2*)(vb + (m * D_STR + kk) * 2);
    float2 b1 = *(const float2*)(vb + (m * D_STR + kk + 2) * 2);
    v2f Br  = {  b0.x,  b1.x };
    v2f Bip = {  b0.y,  b1.y };   // +Im(V^T);  B_actual_im = -Bip (conjugate)
    v2f Bin = { -b0.y, -b1.y };
    // Cre += Ar*Br + Ai*Bip ; Cim += Ai*Br + Ar*Bin
    cre = wmma4(Ar, Br,  cre);
    cre = wmma4(Ai, Bip, cre);
    cim = wmma4(Ai, Br,  cim);
    cim = wmma4(Ar, Bin, cim);
  }
  const int col = m, rb = kk * 8;          // C/D layout: VGPR v -> M = v + 8*half
  for (int v = 0; v < 8; ++v) {
    atomicAdd(ws + WS_VVH_RE + (rb + v) * 16 + col, cre[v]);
    atomicAdd(ws + WS_VVH_IM + (rb + v) * 16 + col, cim[v]);
  }
}

// Per-user: U = (H VVh H^H + sI)^-1 HV, W = (I - U^H HV)^-1, HhU, UW,
// and J += alpha * (HhU W) HhU^H accumulated with WMMA.
// H/V are double-buffered in LDS and prefetched with async global->LDS copies.
__global__ void k_user(const float* __restrict__ Hg, const float* __restrict__ Vg,
                       float* __restrict__ ws, float* __restrict__ outU) {
  __shared__ float2 sVVh[16][16];
  __shared__ float2 sHbuf[2][128];  // 8x16 complex
  __shared__ float2 sVbuf[2][64];   // 16x4 complex
  __shared__ float2 sG[8][16];
  __shared__ float2 sA[8][8];
  __shared__ float2 sRHS[8][4];     // becomes U
  __shared__ float2 sHV[8][4];
  __shared__ float2 sE[4][8];       // augmented [E | I] -> [I | W]
  __shared__ float2 sHhU[16][4];
  __shared__ float2 sM[16][4];
  __shared__ float2 sm[8];
  __shared__ float2 spinv;
  const int lane = threadIdx.x;     // 32 threads = one wave

  float2* fVVh = &sVVh[0][0];
  for (int e = lane; e < 256; e += 32)
    fVVh[e] = make_float2(ws[WS_VVH_RE + e], ws[WS_VVH_IM + e]);

  v8f jre = {0.f,0.f,0.f,0.f,0.f,0.f,0.f,0.f};
  v8f jim = {0.f,0.f,0.f,0.f,0.f,0.f,0.f,0.f};

#if HAVE_ASYNC_LDS
  // Prime the pipeline: 3 async b128 copies (2x H, 1x V) for the first user.
  if (blockIdx.x < I_USERS) {
    const float* hb = Hg + (size_t)blockIdx.x * (R_ANT * T_ANT * 2);
    const float* vb = Vg + (size_t)blockIdx.x * (T_ANT * D_STR * 2);
    async_cp16(&sHbuf[0][lane * 2],        hb + lane * 4);
    async_cp16(&sHbuf[0][(lane + 32) * 2], hb + (lane + 32) * 4);
    async_cp16(&sVbuf[0][lane * 2],        vb + lane * 4);
  }
#endif

  int buf = 0;
  for (int u = blockIdx.x; u < I_USERS; u += gridDim.x, buf ^= 1) {
    __syncthreads();  // drain LDS reads before async engine refills buffers
    const int un = u + (int)gridDim.x;
#if HAVE_ASYNC_LDS
    if (un < I_USERS) {
      const float* hb = Hg + (size_t)un * (R_ANT * T_ANT * 2);
      const float* vb = Vg + (size_t)un * (T_ANT * D_STR * 2);
      float2* Hd = sHbuf[buf ^ 1];
      float2* Vd = sVbuf[buf ^ 1];
      async_cp16(&Hd[lane * 2],        hb + lane * 4);
      async_cp16(&Hd[(lane + 32) * 2], hb + (lane + 32) * 4);
      async_cp16(&Vd[lane * 2],        vb + lane * 4);
      WAIT_ASYNC(3);  // in-order: oldest 3 (current buffer) have landed
    } else {
      WAIT_ASYNC(0);
    }
#else
    {
      const float* hb = Hg + (size_t)u * (R_ANT * T_ANT * 2);
      const float* vb = Vg + (size_t)u * (T_ANT * D_STR * 2);
      float2* fH = sHbuf[buf];
      float2* fV = sVbuf[buf];
      for (int e = lane; e < 128; e += 32) fH[e] = *(const float2*)(hb + 2 * e);
      for (int e = lane; e <  64; e += 32) fV[e] = *(const float2*)(vb + 2 * e);
      if (un < I_USERS) {
        __builtin_prefetch(Hg + (size_t)un * (R_ANT * T_ANT * 2), 0, 3);
        __builtin_prefetch(Vg + (size_t)un * (T_ANT * D_STR * 2), 0, 3);
      }
    }
#endif
    __syncthreads();
    const float2* sHc = sHbuf[buf];   // H[r][k] = sHc[r*16+k]
    const float2* sVc = sVbuf[buf];   // V[t][d] = sVc[t*4+d]

    // G = H * VVh   (8x16)
    for (int e = lane; e < 128; e += 32) {
      int r = e >> 4, t = e & 15;
      float2 acc = make_float2(0.f, 0.f);
      for (int k = 0; k < 16; ++k) acc = cfma_(sHc[r * 16 + k], sVVh[k][t], acc);
      (&sG[0][0])[e] = acc;
    }
    __syncthreads();

    // A = G*H^H + sigma*I   (8x8);  RHS = H*V (8x4)
    for (int e = lane; e < 64; e += 32) {
      int r = e >> 3, q = e & 7;
      float2 acc = make_float2(0.f, 0.f);
      for (int t = 0; t < 16; ++t) acc = cfmac(sG[r][t], sHc[q * 16 + t], acc);
      if (r == q) acc.x += SIGMA_C;
      sA[r][q] = acc;
    }
    {
      int r = lane >> 2, d = lane & 3;
      float2 acc = make_float2(0.f, 0.f);
      for (int k = 0; k < 16; ++k) acc = cfma_(sHc[r * 16 + k], sVc[k * 4 + d], acc);
      sHV[r][d] = acc;
      sRHS[r][d] = acc;
    }
    __syncthreads();

    // Gauss-Jordan: A X = RHS  ->  X = U (8x4)
    for (int k = 0; k < 8; ++k) {
      if (lane < 8)  sm[lane] = sA[lane][k];
      if (lane == 31) spinv = crecip(sA[k][k]);
      __syncthreads();
      if (lane < 12) {
        if (lane < 8) sA[k][lane] = cmul(sA[k][lane], spinv);
        else          sRHS[k][lane - 8] = cmul(sRHS[k][lane - 8], spinv);
      }
      __syncthreads();
      for (int e = lane; e < 96; e += 32) {
        int r = e / 12, c = e % 12;
        if (r != k) {
          if (c < 8) sA[r][c] = csub(sA[r][c], cmul(sm[r], sA[k][c]));
          else       sRHS[r][c - 8] = csub(sRHS[r][c - 8], cmul(sm[r], sRHS[k][c - 8]));
        }
      }
      __syncthreads();
    }

    // Write U out (interleaved complex, layout [I][R][D])
    {
      float2 uv = (&sRHS[0][0])[lane];
      *(float2*)(outU + ((size_t)u * 32 + lane) * 2) = uv;
    }

    // Augmented [E | I],  E = I_D - U^H (H V)
    {
      int d = lane >> 3, c = lane & 7;
      float2 val;
      if (c < 4) {
        float2 acc = make_float2(0.f, 0.f);
        for (int r = 0; r < 8; ++r) acc = cfmca(sRHS[r][d], sHV[r][c], acc);
        val = make_float2((d == c ? 1.f : 0.f) - acc.x, -acc.y);
      } else {
        val = make_float2((d == (c - 4)) ? 1.f : 0.f, 0.f);
      }
      sE[d][c] = val;
    }
    __syncthreads();

    // 4x4 Gauss-Jordan inverse -> W in sE[:,4..7]
    for (int k = 0; k < 4; ++k) {
      if (lane < 4)  sm[lane] = sE[lane][k];
      if (lane == 31) spinv = crecip(sE[k][k]);
      __syncthreads();
      if (lane < 8) sE[k][lane] = cmul(sE[k][lane], spinv);
      __syncthreads();
      {
        int r = lane >> 3, c = lane & 7;
        if (r != k) sE[r][c] = csub(sE[r][c], cmul(sm[r], sE[k][c]));
      }
      __syncthreads();
    }

    // HhU = H^H U (16x4)
    for (int e = lane; e < 64; e += 32) {
      int t = e >> 2, d = e & 3;
      float2 acc = make_float2(0.f, 0.f);
      for (int r = 0; r < 8; ++r) acc = cfmca(sHc[r * 16 + t], sRHS[r][d], acc);
      sHhU[t][d] = acc;
    }
    __syncthreads();
    // M = HhU * W (16x4)
    for (int e = lane; e < 64; e += 32) {
      int t = e >> 2, d = e & 3;
      float2 acc = make_float2(0.f, 0.f);
      for (int k = 0; k < 4; ++k) acc = cfma_(sHhU[t][k], sE[k][4 + d], acc);
      sM[t][d] = acc;
    }
    // UW = U * W (8x4) -> workspace (needed by gradient pass)
    {
      int r = lane >> 2, d = lane & 3;
      float2 acc = make_float2(0.f, 0.f);
      for (int k = 0; k < 4; ++k) acc = cfma_(sRHS[r][k], sE[k][4 + d], acc);
      *(float2*)(ws + WS_UW + ((size_t)u * 32 + lane) * 2) = acc;
    }
    __syncthreads();

    // J += M * HhU^H  via 4 real WMMAs (rank-4 complex outer product)
    {
      int m  = lane & 15;
      int kk = lane >> 4;
      int k0 = kk * 2;
      float2 a0 = sM[m][k0], a1 = sM[m][k0 + 1];
      v2f Ar = { a0.x, a1.x };
      v2f Ai = { a0.y, a1.y };
      float2 b0 = sHhU[m][kk], b1 = sHhU[m][kk + 2];
      v2f Br  = {  b0.x,  b1.x };
      v2f Bip = {  b0.y,  b1.y };
      v2f Bin = { -b0.y, -b1.y };
      jre = wmma4(Ar, Br,  jre);
      jre = wmma4(Ai, Bip, jre);
      jim = wmma4(Ai, Br,  jim);
      jim = wmma4(Ar, Bin, jim);
    }
  }

  // Flush this wave's J contribution
  {
    int col = lane & 15, rb = (lane >> 4) * 8;
    for (int v = 0; v < 8; ++v) {
      atomicAdd(ws + WS_J_RE + (rb + v) * 16 + col, ALPHA_C * jre[v]);
      atomicAdd(ws + WS_J_IM + (rb + v) * 16 + col, ALPHA_C * jim[v]);
    }
  }
}

// grad = J V - alpha H^H (U W);  Vn = V - lr*grad (unscaled); accumulate sum|Vn|^2
__global__ void k_grad(const float* __restrict__ Hg, const float* __restrict__ Vg,
                       float* __restrict__ ws, float* __restrict__ outVn) {
  __shared__ float2 sJ[16][16];
  __shared__ float2 sHbuf[2][128];
  __shared__ float2 sVbuf[2][64];
  __shared__ float2 sUWbuf[2][32];
  const int lane = threadIdx.x;

  float2* fJ = &sJ[0][0];
  for (int e = lane; e < 256; e += 32)
    fJ[e] = make_float2(ws[WS_J_RE + e], ws[WS_J_IM + e]);

#if HAVE_ASYNC_LDS
  if (blockIdx.x < I_USERS) {
    const float* hb  = Hg + (size_t)blockIdx.x * (R_ANT * T_ANT * 2);
    const float* vb  = Vg + (size_t)blockIdx.x * (T_ANT * D_STR * 2);
    const float* uwb = ws + WS_UW + (size_t)blockIdx.x * 64;
    async_cp16(&sHbuf[0][lane * 2],        hb + lane * 4);
    async_cp16(&sHbuf[0][(lane + 32) * 2], hb + (lane + 32) * 4);
    async_cp16(&sVbuf[0][lane * 2],        vb + lane * 4);
    if (lane < 16) async_cp16(&sUWbuf[0][lane * 2], uwb + lane * 4);
  }
#endif

  float psum = 0.f;
  int buf = 0;
  for (int u = blockIdx.x; u < I_USERS; u += gridDim.x, buf ^= 1) {
    __syncthreads();
    const int un = u + (int)gridDim.x;
#if HAVE_ASYNC_LDS
    if (un < I_USERS) {
      const float* hb  = Hg + (size_t)un * (R_ANT * T_ANT * 2);
      const float* vb  = Vg + (size_t)un * (T_ANT * D_STR * 2);
      const float* uwb = ws + WS_UW + (size_t)un * 64;
      float2* Hd  = sHbuf[buf ^ 1];
      float2* Vd  = sVbuf[buf ^ 1];
      float2* UWd = sUWbuf[buf ^ 1];
      async_cp16(&Hd[lane * 2],        hb + lane * 4);
      async_cp16(&Hd[(lane + 32) * 2], hb + (lane + 32) * 4);
      async_cp16(&Vd[lane * 2],        vb + lane * 4);
      if (lane < 16) async_cp16(&UWd[lane * 2], uwb + lane * 4);
      WAIT_ASYNC(4);  // oldest 4 (current buffer) complete
    } else {
      WAIT_ASYNC(0);
    }
#else
    {
      const float* hb  = Hg + (size_t)u * (R_ANT * T_ANT * 2);
      const float* vb  = Vg + (size_t)u * (T_ANT * D_STR * 2);
      float2* fH = sHbuf[buf];
      float2* fV = sVbuf[buf];
      for (int e = lane; e < 128; e += 32) fH[e] = *(const float2*)(hb + 2 * e);
      for (int e = lane; e <  64; e += 32) fV[e] = *(const float2*)(vb + 2 * e);
      sUWbuf[buf][lane] = *(const float2*)(ws + WS_UW + ((size_t)u * 32 + lane) * 2);
      if (un < I_USERS) {
        __builtin_prefetch(Hg + (size_t)un * (R_ANT * T_ANT * 2), 0, 3);
        __builtin_prefetch(Vg + (size_t)un * (T_ANT * D_STR * 2), 0, 3);
      }
    }
#endif
    __syncthreads();
    const float2* sHc  = sHbuf[buf];
    const float2* sVc  = sVbuf[buf];
    const float2* sUWc = sUWbuf[buf];

    for (int e = lane; e < 64; e += 32) {
      int t = e >> 2, d = e & 3;
      float2 g = make_float2(0.f, 0.f);
      for (int s = 0; s < 16; ++s) g = cfma_(sJ[t][s], sVc[s * 4 + d], g);
      float2 h = make_float2(0.f, 0.f);
      for (int r = 0; r < 8; ++r) h = cfmca(sHc[r * 16 + t], sUWc[r * 4 + d], h);
      g.x -= ALPHA_C * h.x;
      g.y -= ALPHA_C * h.y;
      float2 vv = sVc[t * 4 + d];
      float2 vn = make_float2(vv.x - LR_C * g.x, vv.y - LR_C * g.y);
      *(float2*)(outVn + ((size_t)u * 64 + e) * 2) = vn;
      psum = fmaf(vn.x, vn.x, fmaf(vn.y, vn.y, psum));
    }
  }
  // wave32 reduction, one atomic per wave
  for (int off = 16; off > 0; off >>= 1) psum += __shfl_xor(psum, off, 32);
  if (lane == 0) atomicAdd(ws + WS_P, psum);
}

__global__ void k_scale(float* __restrict__ outVn, const float* __restrict__ ws,
                        size_t n) {
  float P = ws[WS_P];
  float s = (P > P_TOT_C) ? sqrtf(P_TOT_C / P) : 1.0f;
  size_t i = (size_t)blockIdx.x * blockDim.x + threadIdx.x;
  size_t stride = (size_t)gridDim.x * blockDim.x;
  for (; i < n; i += stride) outVn[i] *= s;
}

// ------------------------------ launch ---------------------------------------
extern "C" void kernel_launch(void* const* d_in, const int* in_sizes, int n_in,
                              void* d_out, int out_size, void* d_ws, size_t ws_size,
                              hipStream_t stream) {
  (void)in_sizes; (void)n_in; (void)out_size; (void)ws_size;
  const float* Hg = (const float*)d_in[0];   // [I, R, T, 2] f32
  const float* Vg = (const float*)d_in[1];   // [I, T, D, 2] f32
  // d_in[2] = rel_rate (0 -> first branch), d_in[3] = lr (reference uses LR const)
  float* ws   = (float*)d_ws;
  float* outU = (float*)d_out;                                   // [I,R,D] c64
  float* outVn = outU + (size_t)I_USERS * R_ANT * D_STR * 2;     // [I,T,D] c64

  k_init <<<8, 256, 0, stream>>>(ws);
  k_vvh  <<<512, 32, 0, stream>>>(Vg, ws);
  k_user <<<2048, 32, 0, stream>>>(Hg, Vg, ws, outU);
  k_grad <<<2048, 32, 0, stream>>>(Hg, Vg, ws, outVn);
  const size_t nvn = (size_t)I_USERS * T_ANT * D_STR * 2;
  k_scale<<<2048, 256, 0, stream>>>(outVn, ws, nvn);
}